// L1_loss_zero_5712306504420
// MI455X (gfx1250) — compile-verified
//
#include <hip/hip_runtime.h>

// sum(|X - Y| * (Y != 0)) over N f32 elements.
// Pass 1: grid-stride float4 streaming, per-block partial via WMMA-assisted
//         wave reduction -> d_ws[block].
// Pass 2: single block reduces all block partials -> d_out[0]. Deterministic.

typedef __attribute__((ext_vector_type(2))) float v2f;
typedef __attribute__((ext_vector_type(8))) float v8f;

#define BLOCK 256
#define GRID  3072

// Reduce 256 per-thread partials (one per thread, staged through LDS) to a
// single f32. Uses V_WMMA_F32_16X16X4_F32 with B = ones:
//   D[m][n] = sum_k A[m][k] + C[m][n]
// With chained C accumulation over 4 chunks of 64 values, summing D over all
// rows m yields the total of all 256 inputs (mapping of values into A cells
// is irrelevant because B is all-ones). Valid result on all lanes of wave 0.
__device__ __forceinline__ float block_reduce_wmma(float s, float* lds) {
    lds[threadIdx.x] = s;
    __syncthreads();
    float total = 0.0f;
    if (threadIdx.x < 32) {            // wave 0: all 32 lanes active -> EXEC all 1s
        const int lane = threadIdx.x;
        v8f acc = {0.f, 0.f, 0.f, 0.f, 0.f, 0.f, 0.f, 0.f};
        v2f ones = {1.0f, 1.0f};
#pragma unroll
        for (int c = 0; c < 4; ++c) {
            v2f a;
            a[0] = lds[c * 64 + lane];
            a[1] = lds[c * 64 + 32 + lane];
            acc = __builtin_amdgcn_wmma_f32_16x16x4_f32(
                /*neg_a=*/false, a, /*neg_b=*/false, ones,
                /*c_mod=*/(short)0, acc, /*reuse_a=*/false, /*reuse_b=*/false);
        }
        // Lanes 0-15 hold rows 0..7 in acc[0..7]; lanes 16-31 hold rows 8..15.
        float t = acc[0] + acc[1] + acc[2] + acc[3] +
                  acc[4] + acc[5] + acc[6] + acc[7];
        t += __shfl_xor(t, 16, 32);    // fold the two half-wave row groups
        total = t;
    }
    return total;                       // valid on wave-0 lanes
}

__global__ void masked_l1_partial(const float* __restrict__ X,
                                  const float* __restrict__ Y,
                                  float* __restrict__ ws, int n) {
    __shared__ float lds[BLOCK];
    const int tid    = blockIdx.x * blockDim.x + threadIdx.x;
    const int stride = gridDim.x * blockDim.x;
    const int n4     = n >> 2;

    const float4* __restrict__ X4 = reinterpret_cast<const float4*>(X);
    const float4* __restrict__ Y4 = reinterpret_cast<const float4*>(Y);

    float s = 0.0f;
    for (int i = tid; i < n4; i += stride) {
        float4 x = X4[i];
        float4 y = Y4[i];
        s += (y.x != 0.0f) ? fabsf(x.x - y.x) : 0.0f;
        s += (y.y != 0.0f) ? fabsf(x.y - y.y) : 0.0f;
        s += (y.z != 0.0f) ? fabsf(x.z - y.z) : 0.0f;
        s += (y.w != 0.0f) ? fabsf(x.w - y.w) : 0.0f;
    }
    // Scalar tail (n not divisible by 4).
    for (int i = (n4 << 2) + tid; i < n; i += stride) {
        float x = X[i], y = Y[i];
        s += (y != 0.0f) ? fabsf(x - y) : 0.0f;
    }

    float total = block_reduce_wmma(s, lds);
    if (threadIdx.x == 0) ws[blockIdx.x] = total;
}

__global__ void final_reduce(const float* __restrict__ ws, int nparts,
                             float* __restrict__ out) {
    __shared__ float lds[BLOCK];
    float s = 0.0f;
    for (int i = threadIdx.x; i < nparts; i += BLOCK) s += ws[i];
    float total = block_reduce_wmma(s, lds);
    if (threadIdx.x == 0) out[0] = total;
}

extern "C" void kernel_launch(void* const* d_in, const int* in_sizes, int n_in,
                              void* d_out, int out_size, void* d_ws, size_t ws_size,
                              hipStream_t stream) {
    (void)n_in; (void)out_size; (void)ws_size;
    const float* X = (const float*)d_in[0];
    const float* Y = (const float*)d_in[1];
    float* out     = (float*)d_out;
    float* ws      = (float*)d_ws;     // GRID floats = 12 KB of scratch
    const int n    = in_sizes[0];

    masked_l1_partial<<<GRID, BLOCK, 0, stream>>>(X, Y, ws, n);
    final_reduce<<<1, BLOCK, 0, stream>>>(ws, GRID, out);
}